// MPN_31190052503579
// MI455X (gfx1250) — compile-verified
//
#include <hip/hip_runtime.h>

// ---------------------------------------------------------------------------
// DimeNet-style interaction + output block for MI455X (gfx1250, wave32).
// All dense H x H (and K=1024 bilinear) GEMMs run on V_WMMA_F32_16X16X4_F32.
// Weights are pre-packed into fragment-native layout so every B-fragment is a
// single coalesced global_load_b64. The bilinear kernel processes two 16-row
// tiles per wave so each B-fragment feeds two WMMAs.
// Scratch: 3*E*H + T*8 + 15*16384 + N*H + G*H + G floats (~325 MB here).
// ---------------------------------------------------------------------------

typedef float v2f __attribute__((ext_vector_type(2)));
typedef float v8f __attribute__((ext_vector_type(8)));

#define H   128
#define LDA 132        // padded LDS row stride -> conflict-free A-frag reads
#define PKW 16384      // floats per packed 128x128 weight

__device__ __forceinline__ float silu_f(float v) {
    return v / (1.0f + __expf(-v));
}

// Packed weight layout, float2 units:
//   pk[kk*256 + nt*32 + lane] = { W[kb][col], W[kb+1][col] }
//   kb = kk*4 + (lane&16 ? 2 : 0), col = nt*16 + (lane&15)
// This matches the 16x16x4 f32 B-fragment exactly: one b64 load per WMMA.
__global__ void repack_weight_kernel(const float* __restrict__ src,
                                     float2* __restrict__ dst,
                                     int base, int kstride, int nstride)
{
    const int t = blockIdx.x * blockDim.x + threadIdx.x;   // 0..8191
    if (t >= 32 * 8 * 32) return;
    const int lane = t & 31;
    const int nt   = (t >> 5) & 7;
    const int kk   = t >> 8;
    const int kb   = kk * 4 + ((lane & 16) ? 2 : 0);
    const int col  = nt * 16 + (lane & 15);
    float2 v;
    v.x = src[(size_t)base + (size_t)kb * kstride + (size_t)col * nstride];
    v.y = src[(size_t)base + (size_t)(kb + 1) * kstride + (size_t)col * nstride];
    dst[t] = v;
}

// One wave computes D[16,128] = A[16,128] * W[128,128] with f32 WMMA.
// As : LDS tile (stride LDA).  Wp : packed weight (float2, layout above).
__device__ __forceinline__ void wave_gemm_16x128(const float* As,
                                                 const float2* __restrict__ Wp,
                                                 v8f acc[8], int lane)
{
    const int row  = lane & 15;
    const int koff = (lane & 16) ? 2 : 0;
    for (int kk = 0; kk < 32; ++kk) {
        const int kb = kk * 4 + koff;
        v2f a;
        a.x = As[row * LDA + kb];
        a.y = As[row * LDA + kb + 1];
        const float2* wp = Wp + (size_t)kk * 256 + lane;
        #pragma unroll
        for (int nt = 0; nt < 8; ++nt) {
            const float2 b2 = wp[nt * 32];
            v2f b; b.x = b2.x; b.y = b2.y;
            acc[nt] = __builtin_amdgcn_wmma_f32_16x16x4_f32(
                false, a, false, b, (short)0, acc[nt], false, false);
        }
    }
}

// ---------------------------------------------------------------------------
// Kernel 1: x_kj = silu(x@W_kj + b_kj) * (rbf@W_rbf_i);  x_ji = silu(x@W_ji+b)
// ---------------------------------------------------------------------------
__global__ void __launch_bounds__(128)
edge_mlp_kernel(const float* __restrict__ x, const float* __restrict__ rbf,
                const float2* __restrict__ Wkj_pk, const float* __restrict__ b_kj,
                const float2* __restrict__ Wji_pk, const float* __restrict__ b_ji,
                const float* __restrict__ W_rbf_i,
                float* __restrict__ xkj, float* __restrict__ xji, int E)
{
    __shared__ float sA[4][16 * LDA];
    __shared__ float sRbf[4][16 * 8];
    __shared__ float sWr[6 * H];
    const int lane = threadIdx.x & 31;
    const int wave = threadIdx.x >> 5;

    for (int i = threadIdx.x; i < 6 * H; i += (int)blockDim.x) sWr[i] = W_rbf_i[i];
    __syncthreads();

    const int e0 = (blockIdx.x * 4 + wave) * 16;
    if (e0 >= E) return;

    for (int r = 0; r < 16; ++r) {
        const int e = min(e0 + r, E - 1);
        float4 v = ((const float4*)(x + (size_t)e * H))[lane];
        float* d = &sA[wave][r * LDA + lane * 4];
        d[0] = v.x; d[1] = v.y; d[2] = v.z; d[3] = v.w;
    }
    if (lane < 16) {
        const int e = min(e0 + lane, E - 1);
        #pragma unroll
        for (int q = 0; q < 6; ++q) sRbf[wave][lane * 8 + q] = rbf[(size_t)e * 6 + q];
    }

    const int col = lane & 15;
    const int rb  = (lane & 16) ? 8 : 0;

    // ---- x_kj ----
    {
        v8f acc[8] = {};
        wave_gemm_16x128(sA[wave], Wkj_pk, acc, lane);
        #pragma unroll
        for (int nt = 0; nt < 8; ++nt) {
            const int n    = nt * 16 + col;
            const float bn = b_kj[n];
            #pragma unroll
            for (int r8 = 0; r8 < 8; ++r8) {
                const int row = rb + r8;
                float rh = 0.0f;
                #pragma unroll
                for (int q = 0; q < 6; ++q)
                    rh += sRbf[wave][row * 8 + q] * sWr[q * H + n];
                const float v = silu_f(acc[nt][r8] + bn) * rh;
                const int e = e0 + row;
                if (e < E) xkj[(size_t)e * H + n] = v;
            }
        }
    }
    // ---- x_ji ----
    {
        v8f acc[8] = {};
        wave_gemm_16x128(sA[wave], Wji_pk, acc, lane);
        #pragma unroll
        for (int nt = 0; nt < 8; ++nt) {
            const int n    = nt * 16 + col;
            const float bn = b_ji[n];
            #pragma unroll
            for (int r8 = 0; r8 < 8; ++r8) {
                const int e = e0 + rb + r8;
                if (e < E) xji[(size_t)e * H + n] = silu_f(acc[nt][r8] + bn);
            }
        }
    }
}

// ---------------------------------------------------------------------------
// Kernel 2: sbf_b = sbf @ W_sbf   [T,42]@[42,8]
// ---------------------------------------------------------------------------
__global__ void sbfb_kernel(const float* __restrict__ sbf,
                            const float* __restrict__ W_sbf,
                            float* __restrict__ sbfb, int T)
{
    __shared__ float sW[42 * 8];
    for (int i = threadIdx.x; i < 336; i += (int)blockDim.x) sW[i] = W_sbf[i];
    __syncthreads();
    const int w = blockIdx.x * blockDim.x + threadIdx.x;
    if (w >= T) return;
    float r[42];
    #pragma unroll
    for (int q = 0; q < 42; ++q) r[q] = sbf[(size_t)w * 42 + q];
    #pragma unroll
    for (int j = 0; j < 8; ++j) {
        float s = 0.0f;
        #pragma unroll
        for (int q = 0; q < 42; ++q) s += r[q] * sW[q * 8 + j];
        sbfb[(size_t)w * 8 + j] = s;
    }
}

// ---------------------------------------------------------------------------
// Kernel 3: bilinear einsum as GEMM with on-the-fly outer-product A,
//           K = 1024 (8 j-slices x 128), scatter-add via idx_ji into agg.
// xk[w,i] = sum_{j,l} sbf_b[w,j] * m[w,l] * W_bil[i,j,l],  m = x_kj[idx_kj]
// Two 16-row tiles per wave: each B-fragment load feeds two WMMAs.
// ---------------------------------------------------------------------------
__global__ void __launch_bounds__(128)
bilinear_kernel(const float* __restrict__ xkj, const float* __restrict__ sbfb,
                const float2* __restrict__ Wbil_pk, const int* __restrict__ idx_kj,
                const int* __restrict__ idx_ji, float* __restrict__ agg,
                int T, int E)
{
    __shared__ float sM[4][32 * LDA];
    __shared__ float sS[4][32 * 8];
    __shared__ int   sJ[4][32];
    const int lane = threadIdx.x & 31;
    const int wave = threadIdx.x >> 5;
    const int w0   = (blockIdx.x * 4 + wave) * 32;
    if (w0 >= T) return;

    sJ[wave][lane] = idx_ji[min(w0 + lane, T - 1)];
    {   // sbf_b: 32 rows x 8, one row (two float4) per lane
        const int w = min(w0 + lane, T - 1);
        float4 v0 = *(const float4*)(sbfb + (size_t)w * 8);
        float4 v1 = *(const float4*)(sbfb + (size_t)w * 8 + 4);
        float* d = &sS[wave][lane * 8];
        d[0] = v0.x; d[1] = v0.y; d[2] = v0.z; d[3] = v0.w;
        d[4] = v1.x; d[5] = v1.y; d[6] = v1.z; d[7] = v1.w;
    }
    for (int r = 0; r < 32; ++r) {
        const int e = idx_kj[min(w0 + r, T - 1)];
        float4 v = ((const float4*)(xkj + (size_t)e * H))[lane];
        float* d = &sM[wave][r * LDA + lane * 4];
        d[0] = v.x; d[1] = v.y; d[2] = v.z; d[3] = v.w;
    }

    const int row  = lane & 15;
    const int koff = (lane & 16) ? 2 : 0;
    const int col  = lane & 15;
    v8f acc0[8] = {};
    v8f acc1[8] = {};
    for (int j = 0; j < 8; ++j) {       // j outer: 64KB packed slice stays hot in WGP$
        const float s0 = sS[wave][row * 8 + j];
        const float s1 = sS[wave][(16 + row) * 8 + j];
        const float2* Wj = Wbil_pk + (size_t)j * (PKW / 2);
        for (int kk = 0; kk < 32; ++kk) {
            const int l0 = kk * 4 + koff;
            v2f a0, a1;
            a0.x = s0 * sM[wave][row * LDA + l0];
            a0.y = s0 * sM[wave][row * LDA + l0 + 1];
            a1.x = s1 * sM[wave][(16 + row) * LDA + l0];
            a1.y = s1 * sM[wave][(16 + row) * LDA + l0 + 1];
            const float2* wp = Wj + (size_t)kk * 256 + lane;
            #pragma unroll
            for (int nt = 0; nt < 8; ++nt) {
                const float2 b2 = wp[nt * 32];
                v2f b; b.x = b2.x; b.y = b2.y;
                acc0[nt] = __builtin_amdgcn_wmma_f32_16x16x4_f32(
                    false, a0, false, b, (short)0, acc0[nt], false, false);
                acc1[nt] = __builtin_amdgcn_wmma_f32_16x16x4_f32(
                    false, a1, false, b, (short)0, acc1[nt], false, false);
            }
        }
    }
    const int rb = (lane & 16) ? 8 : 0;
    #pragma unroll
    for (int r8 = 0; r8 < 8; ++r8) {
        const int w = w0 + rb + r8;
        if (w < T) {
            const int e = sJ[wave][rb + r8];
            #pragma unroll
            for (int nt = 0; nt < 8; ++nt)
                atomicAdd(&agg[(size_t)e * H + nt * 16 + col], acc0[nt][r8]);
        }
    }
    #pragma unroll
    for (int r8 = 0; r8 < 8; ++r8) {
        const int w = w0 + 16 + rb + r8;
        if (w < T) {
            const int e = sJ[wave][16 + rb + r8];
            #pragma unroll
            for (int nt = 0; nt < 8; ++nt)
                atomicAdd(&agg[(size_t)e * H + nt * 16 + col], acc1[nt][r8]);
        }
    }
}

// ---------------------------------------------------------------------------
// Kernel 4: h = silu((x_ji+agg)@W_lin + b) * (rbf@W_rbf_o); hn[i_idx] += h
// ---------------------------------------------------------------------------
__global__ void __launch_bounds__(128)
edge_out_kernel(const float* __restrict__ xji, const float* __restrict__ agg,
                const float* __restrict__ rbf, const float2* __restrict__ Wlin_pk,
                const float* __restrict__ b_lin, const float* __restrict__ W_rbf_o,
                const int* __restrict__ i_idx, float* __restrict__ hn, int E)
{
    __shared__ float sA[4][16 * LDA];
    __shared__ float sRbf[4][16 * 8];
    __shared__ float sWr[6 * H];
    __shared__ int   sI[4][16];
    const int lane = threadIdx.x & 31;
    const int wave = threadIdx.x >> 5;

    for (int i = threadIdx.x; i < 6 * H; i += (int)blockDim.x) sWr[i] = W_rbf_o[i];
    __syncthreads();

    const int e0 = (blockIdx.x * 4 + wave) * 16;
    if (e0 >= E) return;

    for (int r = 0; r < 16; ++r) {
        const int e = min(e0 + r, E - 1);
        float4 a = ((const float4*)(xji + (size_t)e * H))[lane];
        float4 g = ((const float4*)(agg + (size_t)e * H))[lane];
        float* d = &sA[wave][r * LDA + lane * 4];
        d[0] = a.x + g.x; d[1] = a.y + g.y; d[2] = a.z + g.z; d[3] = a.w + g.w;
    }
    if (lane < 16) {
        const int e = min(e0 + lane, E - 1);
        sI[wave][lane] = i_idx[e];
        #pragma unroll
        for (int q = 0; q < 6; ++q) sRbf[wave][lane * 8 + q] = rbf[(size_t)e * 6 + q];
    }

    const int col = lane & 15;
    const int rb  = (lane & 16) ? 8 : 0;
    v8f acc[8] = {};
    wave_gemm_16x128(sA[wave], Wlin_pk, acc, lane);
    #pragma unroll
    for (int nt = 0; nt < 8; ++nt) {
        const int n    = nt * 16 + col;
        const float bn = b_lin[n];
        #pragma unroll
        for (int r8 = 0; r8 < 8; ++r8) {
            const int row = rb + r8;
            float rh = 0.0f;
            #pragma unroll
            for (int q = 0; q < 6; ++q)
                rh += sRbf[wave][row * 8 + q] * sWr[q * H + n];
            const float v = silu_f(acc[nt][r8] + bn) * rh;
            const int e = e0 + row;
            if (e < E) atomicAdd(&hn[(size_t)sI[wave][row] * H + n], v);
        }
    }
}

// ---------------------------------------------------------------------------
// Kernel 5: node MLP: 3x (silu(hn@W+b)) then @W_out; pool[batch] += out
// ---------------------------------------------------------------------------
__global__ void __launch_bounds__(128)
node_mlp_kernel(const float* __restrict__ hn,
                const float2* __restrict__ W0pk, const float* __restrict__ b0v,
                const float2* __restrict__ W1pk, const float* __restrict__ b1v,
                const float2* __restrict__ W2pk, const float* __restrict__ b2v,
                const float2* __restrict__ Woutpk, const int* __restrict__ batch,
                float* __restrict__ pool, int N)
{
    __shared__ float sA[4][16 * LDA];
    const int lane = threadIdx.x & 31;
    const int wave = threadIdx.x >> 5;
    const int n0   = (blockIdx.x * 4 + wave) * 16;
    if (n0 >= N) return;

    for (int r = 0; r < 16; ++r) {
        const int nn = min(n0 + r, N - 1);
        float4 v = ((const float4*)(hn + (size_t)nn * H))[lane];
        float* d = &sA[wave][r * LDA + lane * 4];
        d[0] = v.x; d[1] = v.y; d[2] = v.z; d[3] = v.w;
    }

    const int col = lane & 15;
    const int rb  = (lane & 16) ? 8 : 0;
    const float2* Ws[3] = { W0pk, W1pk, W2pk };
    const float*  bs[3] = { b0v, b1v, b2v };

    for (int L = 0; L < 3; ++L) {
        v8f acc[8] = {};
        wave_gemm_16x128(sA[wave], Ws[L], acc, lane);
        #pragma unroll
        for (int nt = 0; nt < 8; ++nt) {
            const float bn = bs[L][nt * 16 + col];
            #pragma unroll
            for (int r8 = 0; r8 < 8; ++r8)
                sA[wave][(rb + r8) * LDA + nt * 16 + col] = silu_f(acc[nt][r8] + bn);
        }
    }

    v8f acc[8] = {};
    wave_gemm_16x128(sA[wave], Woutpk, acc, lane);
    #pragma unroll
    for (int r8 = 0; r8 < 8; ++r8) {
        const int nn = n0 + rb + r8;
        if (nn < N) {
            const int g = batch[nn];
            #pragma unroll
            for (int nt = 0; nt < 8; ++nt)
                atomicAdd(&pool[(size_t)g * H + nt * 16 + col], acc[nt][r8]);
        }
    }
}

__global__ void count_kernel(const int* __restrict__ batch,
                             float* __restrict__ cnt, int N)
{
    const int v = blockIdx.x * blockDim.x + threadIdx.x;
    if (v < N) atomicAdd(&cnt[batch[v]], 1.0f);
}

__global__ void finalize_kernel(const float* __restrict__ pool,
                                const float* __restrict__ cnt,
                                float* __restrict__ out, int total)
{
    const int i = blockIdx.x * blockDim.x + threadIdx.x;
    if (i >= total) return;
    out[i] = pool[i] / fmaxf(cnt[i / H], 1.0f);
}

// ---------------------------------------------------------------------------
extern "C" void kernel_launch(void* const* d_in, const int* in_sizes, int n_in,
                              void* d_out, int out_size, void* d_ws, size_t ws_size,
                              hipStream_t stream)
{
    const float* x       = (const float*)d_in[0];
    const float* rbf     = (const float*)d_in[1];
    const float* sbf     = (const float*)d_in[2];
    const float* W_rbf_i = (const float*)d_in[3];
    const float* W_sbf   = (const float*)d_in[4];
    const float* W_kj    = (const float*)d_in[5];
    const float* b_kj    = (const float*)d_in[6];
    const float* W_ji    = (const float*)d_in[7];
    const float* b_ji    = (const float*)d_in[8];
    const float* W_bil   = (const float*)d_in[9];
    const float* W_lin   = (const float*)d_in[10];
    const float* b_lin   = (const float*)d_in[11];
    const float* W_rbf_o = (const float*)d_in[12];
    const float* W_o0    = (const float*)d_in[13];
    const float* b_o0    = (const float*)d_in[14];
    const float* W_o1    = (const float*)d_in[15];
    const float* b_o1    = (const float*)d_in[16];
    const float* W_o2    = (const float*)d_in[17];
    const float* b_o2    = (const float*)d_in[18];
    const float* W_out   = (const float*)d_in[19];
    const int*   idx_kj  = (const int*)d_in[20];
    const int*   idx_ji  = (const int*)d_in[21];
    const int*   i_idx   = (const int*)d_in[22];
    const int*   batch   = (const int*)d_in[23];

    const int E = in_sizes[0] / H;
    const int T = in_sizes[20];
    const int N = in_sizes[23];
    const int G = out_size / H;

    float* ws   = (float*)d_ws;
    float* xkj  = ws;  ws += (size_t)E * H;
    float* xji  = ws;  ws += (size_t)E * H;
    float* agg  = ws;  ws += (size_t)E * H;
    float* sbfb = ws;  ws += (size_t)T * 8;
    float* pk   = ws;  ws += (size_t)15 * PKW;   // 7 dense + 8 bilinear slices
    float* hnb  = ws;  ws += (size_t)N * H;
    float* pool = ws;  ws += (size_t)G * H;
    float* cnt  = ws;  ws += G;

    float2* pk_kj  = (float2*)(pk + 0 * PKW);
    float2* pk_ji  = (float2*)(pk + 1 * PKW);
    float2* pk_lin = (float2*)(pk + 2 * PKW);
    float2* pk_o0  = (float2*)(pk + 3 * PKW);
    float2* pk_o1  = (float2*)(pk + 4 * PKW);
    float2* pk_o2  = (float2*)(pk + 5 * PKW);
    float2* pk_out = (float2*)(pk + 6 * PKW);
    float2* pk_bil = (float2*)(pk + 7 * PKW);

    hipMemsetAsync(agg,  0, (size_t)E * H * sizeof(float), stream);
    hipMemsetAsync(hnb,  0, (size_t)N * H * sizeof(float), stream);
    hipMemsetAsync(pool, 0, (size_t)G * H * sizeof(float), stream);
    hipMemsetAsync(cnt,  0, (size_t)G * sizeof(float), stream);

    const int rpThreads = 32 * 8 * 32;
    const int rpBlocks  = (rpThreads + 255) / 256;
    // Plain [k][n] weights (kstride=H, nstride=1)
    repack_weight_kernel<<<rpBlocks, 256, 0, stream>>>(W_kj,  pk_kj,  0, H, 1);
    repack_weight_kernel<<<rpBlocks, 256, 0, stream>>>(W_ji,  pk_ji,  0, H, 1);
    repack_weight_kernel<<<rpBlocks, 256, 0, stream>>>(W_lin, pk_lin, 0, H, 1);
    repack_weight_kernel<<<rpBlocks, 256, 0, stream>>>(W_o0,  pk_o0,  0, H, 1);
    repack_weight_kernel<<<rpBlocks, 256, 0, stream>>>(W_o1,  pk_o1,  0, H, 1);
    repack_weight_kernel<<<rpBlocks, 256, 0, stream>>>(W_o2,  pk_o2,  0, H, 1);
    repack_weight_kernel<<<rpBlocks, 256, 0, stream>>>(W_out, pk_out, 0, H, 1);
    // W_bil [i][j][l]: slice j -> W[k=l][n=i] = Wb[i*1024 + j*128 + l]
    for (int j = 0; j < 8; ++j)
        repack_weight_kernel<<<rpBlocks, 256, 0, stream>>>(
            W_bil, pk_bil + (size_t)j * (PKW / 2), j * H, 1, 1024);

    sbfb_kernel<<<(T + 127) / 128, 128, 0, stream>>>(sbf, W_sbf, sbfb, T);

    const int etiles = (E + 15) / 16;
    edge_mlp_kernel<<<(etiles + 3) / 4, 128, 0, stream>>>(
        x, rbf, pk_kj, b_kj, pk_ji, b_ji, W_rbf_i, xkj, xji, E);

    const int t32tiles = (T + 31) / 32;
    bilinear_kernel<<<(t32tiles + 3) / 4, 128, 0, stream>>>(
        xkj, sbfb, pk_bil, idx_kj, idx_ji, agg, T, E);

    edge_out_kernel<<<(etiles + 3) / 4, 128, 0, stream>>>(
        xji, agg, rbf, pk_lin, b_lin, W_rbf_o, i_idx, hnb, E);

    count_kernel<<<(N + 255) / 256, 256, 0, stream>>>(batch, cnt, N);

    const int ntiles = (N + 15) / 16;
    node_mlp_kernel<<<(ntiles + 3) / 4, 128, 0, stream>>>(
        hnb, pk_o0, b_o0, pk_o1, b_o1, pk_o2, b_o2, pk_out, batch, pool, N);

    finalize_kernel<<<(G * H + 255) / 256, 256, 0, stream>>>(
        pool, cnt, (float*)d_out, G * H);
}